// QCNN_Diff_29978871726317
// MI455X (gfx1250) — compile-verified
//
#include <hip/hip_runtime.h>
#include <hip/hip_bf16.h>

// ---------------------------------------------------------------------------
// QCNN on MI455X (gfx1250, wave32).
//  Phase 1: init circuit (per-batch angles) simulated one-wave-per-state,
//           8 complex amps per lane, qubits 3..7 via shfl_xor.
//  Phase 2: branch circuits (shared angles) applied to 256 basis states to
//           materialize each branch's unitary, stored TRANSPOSED: KT[a][j]
//           so WMMA B-fragments are per-lane contiguous b64 loads.
//  Phase 3: psi_out = psi0 x K as complex f32 GEMM with V_WMMA_F32_16X16X4_F32,
//           32 rows per wave (2 D-tiles share every B fragment),
//           fused signed-|C|^2 reduction for <Z3>,<Z7>.
//  Phase 4: tiny MLP.
// ---------------------------------------------------------------------------

typedef __attribute__((ext_vector_type(2))) float v2f;
typedef __attribute__((ext_vector_type(8))) float v8f;

#define DIM 256  // 2^8 amplitudes

struct cplx { float re, im; };
static __device__ __forceinline__ cplx cmul(cplx a, cplx b) {
    return { a.re * b.re - a.im * b.im, a.re * b.im + a.im * b.re };
}
static __device__ __forceinline__ cplx cadd(cplx a, cplx b) {
    return { a.re + b.re, a.im + b.im };
}
static __device__ __forceinline__ cplx cshfl_xor(cplx v, int mask) {
    cplx r;
    r.re = __shfl_xor(v.re, mask, 32);
    r.im = __shfl_xor(v.im, mask, 32);
    return r;
}

struct Gate { cplx u00, u01, u10, u11; };

static __device__ __forceinline__ Gate gate_ry(float t) {
    float c = cosf(0.5f * t), s = sinf(0.5f * t);
    return { {c, 0.f}, {-s, 0.f}, {s, 0.f}, {c, 0.f} };
}
static __device__ __forceinline__ Gate gate_rz(float t) {
    float c = cosf(0.5f * t), s = sinf(0.5f * t);
    return { {c, -s}, {0.f, 0.f}, {0.f, 0.f}, {c, s} };
}
static __device__ __forceinline__ Gate gate_rx(float t) {
    float c = cosf(0.5f * t), s = sinf(0.5f * t);
    return { {c, 0.f}, {0.f, -s}, {0.f, -s}, {c, 0.f} };
}
static __device__ __forceinline__ Gate gate_u3(float th, float ph, float lm) {
    float c = cosf(0.5f * th), s = sinf(0.5f * th);
    return { {c, 0.f},
             {-cosf(lm) * s, -sinf(lm) * s},
             { cosf(ph) * s,  sinf(ph) * s},
             { cosf(ph + lm) * c, sinf(ph + lm) * c} };
}
static __device__ __forceinline__ Gate gate_x() {
    return { {0.f, 0.f}, {1.f, 0.f}, {1.f, 0.f}, {0.f, 0.f} };
}

// amplitude index a = (lane << 3) | k ; qubit w bit: w<3 -> k bit, w>=3 -> lane bit
static __device__ void apply1q(cplx s[8], int lane, int w, Gate g) {
    if (w < 3) {
        int m = 1 << w;
        #pragma unroll
        for (int k = 0; k < 8; ++k) {
            if (k & m) continue;
            cplx a0 = s[k], a1 = s[k | m];
            s[k]     = cadd(cmul(g.u00, a0), cmul(g.u01, a1));
            s[k | m] = cadd(cmul(g.u10, a0), cmul(g.u11, a1));
        }
    } else {
        int mask  = 1 << (w - 3);
        int mybit = (lane >> (w - 3)) & 1;
        #pragma unroll
        for (int k = 0; k < 8; ++k) {
            cplx t = cshfl_xor(s[k], mask);   // partner amplitude (all lanes active)
            cplx r;
            if (mybit == 0) r = cadd(cmul(g.u00, s[k]), cmul(g.u01, t));
            else            r = cadd(cmul(g.u10, t),    cmul(g.u11, s[k]));
            s[k] = r;
        }
    }
}

static __device__ void applyCtrl(cplx s[8], int lane, int c, int t, Gate g) {
    if (t < 3) {
        int m = 1 << t;
        #pragma unroll
        for (int k = 0; k < 8; ++k) {
            if (k & m) continue;
            int ctrl = (c < 3) ? ((k >> c) & 1) : ((lane >> (c - 3)) & 1);
            if (!ctrl) continue;              // no cross-lane ops here: safe divergence
            cplx a0 = s[k], a1 = s[k | m];
            s[k]     = cadd(cmul(g.u00, a0), cmul(g.u01, a1));
            s[k | m] = cadd(cmul(g.u10, a0), cmul(g.u11, a1));
        }
    } else {
        int mask  = 1 << (t - 3);
        int mybit = (lane >> (t - 3)) & 1;
        #pragma unroll
        for (int k = 0; k < 8; ++k) {
            cplx tv = cshfl_xor(s[k], mask);  // shuffle BEFORE predication
            int ctrl = (c < 3) ? ((k >> c) & 1) : ((lane >> (c - 3)) & 1);
            cplx r;
            if (mybit == 0) r = cadd(cmul(g.u00, s[k]), cmul(g.u01, tv));
            else            r = cadd(cmul(g.u10, tv),   cmul(g.u11, s[k]));
            s[k] = ctrl ? r : s[k];
        }
    }
}

// --------------------------- Phase 1: init_state ---------------------------
__global__ void init_kernel(const float* __restrict__ theta,
                            const float* __restrict__ phi,
                            float* __restrict__ Ar, float* __restrict__ Ai,
                            int B) {
    int wid  = blockIdx.x * (blockDim.x >> 5) + (threadIdx.x >> 5);  // batch index
    int lane = threadIdx.x & 31;
    if (wid >= B) return;

    cplx s[8];
    #pragma unroll
    for (int k = 0; k < 8; ++k) s[k] = {0.f, 0.f};
    if (lane == 0) s[0] = {1.f, 0.f};

    Gate Ry = gate_ry(theta[wid]);
    Gate Rz = gate_rz(phi[wid]);
    Gate X  = gate_x();

    for (int cyc = 0; cyc < 4; ++cyc) {
        for (int w = 0; w < 8; ++w) {
            apply1q(s, lane, w, Ry);
            apply1q(s, lane, w, Rz);
        }
        for (int w = 0; w < 8; ++w)
            applyCtrl(s, lane, w, (w + 1) & 7, X);
    }

    #pragma unroll
    for (int k = 0; k < 8; ++k) {
        int a = (lane << 3) | k;
        Ar[(size_t)wid * DIM + a] = s[k].re;
        Ai[(size_t)wid * DIM + a] = s[k].im;
    }
}

// ---------------- Phase 2: branch unitaries, stored as KT[a][j] ------------
__global__ void branch_kernel(const float* __restrict__ ax, const float* __restrict__ ay,
                              const float* __restrict__ az, const float* __restrict__ u3x,
                              const float* __restrict__ u3y, const float* __restrict__ u3z,
                              float* __restrict__ Kbase) {
    int wid  = blockIdx.x * (blockDim.x >> 5) + (threadIdx.x >> 5); // 0..767
    int lane = threadIdx.x & 31;
    int br = wid >> 8;         // 0:rx 1:ry 2:rz
    int j  = wid & 255;        // basis state index

    const float* ang = (br == 0) ? ax : (br == 1) ? ay : az;
    const float* up  = (br == 0) ? u3x : (br == 1) ? u3y : u3z;

    cplx s[8];
    #pragma unroll
    for (int k = 0; k < 8; ++k) {
        int a = (lane << 3) | k;
        s[k] = (a == j) ? cplx{1.f, 0.f} : cplx{0.f, 0.f};
    }

    const int pa_c[7] = {0, 2, 4, 6, 1, 3, 5};
    const int pa_t[7] = {1, 3, 5, 7, 2, 4, 6};
    const int pb_c[3] = {1, 5, 3};
    const int pb_t[3] = {3, 7, 5};
    const int w1[4]   = {1, 3, 5, 7};
    const int w2[2]   = {3, 7};

    for (int i = 0; i < 7; ++i) {
        float t = ang[i];
        Gate g = (br == 0) ? gate_rx(t) : (br == 1) ? gate_ry(t) : gate_rz(t);
        applyCtrl(s, lane, pa_c[i], pa_t[i], g);
    }
    for (int i = 0; i < 4; ++i) {
        Gate g = gate_u3(up[i * 3 + 0], up[i * 3 + 1], up[i * 3 + 2]);
        apply1q(s, lane, w1[i], g);
    }
    for (int i = 0; i < 3; ++i) {
        float t = ang[7 + i];
        Gate g = (br == 0) ? gate_rx(t) : (br == 1) ? gate_ry(t) : gate_rz(t);
        applyCtrl(s, lane, pb_c[i], pb_t[i], g);
    }
    for (int i = 0; i < 2; ++i) {
        Gate g = gate_u3(up[(4 + i) * 3 + 0], up[(4 + i) * 3 + 1], up[(4 + i) * 3 + 2]);
        apply1q(s, lane, w2[i], g);
    }

    // KT[a][j] = (circuit e_j)[a] : psi_out[b][a] = sum_j psi0[b][j] * KT[a][j]
    // Transposed layout makes the WMMA B-fragment (K[krow..krow+1][ncol]) a
    // contiguous per-lane b64 load in phase 3.
    float* KTr = Kbase + (size_t)br * 2 * DIM * DIM;
    float* KTi = KTr + DIM * DIM;
    #pragma unroll
    for (int k = 0; k < 8; ++k) {
        int a = (lane << 3) | k;
        KTr[(size_t)a * DIM + j] = s[k].re;
        KTi[(size_t)a * DIM + j] = s[k].im;
    }
}

// ----------- Phase 3: complex GEMM via f32 WMMA + fused <Z> readout --------
#define ASTRIDE 268   // padded LDS row stride (floats); 1072B = 16B-aligned, no conflicts
#define ROWS 32       // rows (batch states) per wave: 2 D-tiles share each B fragment

__global__ __launch_bounds__(32)
void gemm_expz_kernel(const float* __restrict__ Ar, const float* __restrict__ Ai,
                      const float* __restrict__ Kbase, float* __restrict__ feats,
                      int rowBlocks) {
    __shared__ __align__(16) float lAr[ROWS * ASTRIDE];
    __shared__ __align__(16) float lAi[ROWS * ASTRIDE];

    int br = blockIdx.x / rowBlocks;          // branch 0..2
    int rb = blockIdx.x % rowBlocks;          // 32-row block
    int r0 = rb * ROWS;
    int lane = threadIdx.x;                   // 0..31 (one wave)
    int half = lane >> 4;                     // D rows r (half=0) / r+8 (half=1)
    int l15  = lane & 15;

    // stage ROWSx256 re/im psi0 block into LDS with b128 loads
    for (int idx = lane; idx < ROWS * (DIM / 4); idx += 32) {
        int row = idx >> 6, f4 = (idx & 63) << 2;
        *(float4*)&lAr[row * ASTRIDE + f4] =
            *(const float4*)&Ar[(size_t)(r0 + row) * DIM + f4];
        *(float4*)&lAi[row * ASTRIDE + f4] =
            *(const float4*)&Ai[(size_t)(r0 + row) * DIM + f4];
    }
    __syncthreads();

    const float* KTr = Kbase + (size_t)br * 2 * DIM * DIM;
    const float* KTi = KTr + DIM * DIM;

    float acc3[2][8], acc7[2][8];
    #pragma unroll
    for (int tfrag = 0; tfrag < 2; ++tfrag)
        #pragma unroll
        for (int r = 0; r < 8; ++r) { acc3[tfrag][r] = 0.f; acc7[tfrag][r] = 0.f; }

    // sign of bit3 of column index depends only on (lane & 8); bit7 only on tile
    float s3 = (lane & 8) ? -1.f : 1.f;

    for (int nt = 0; nt < 16; ++nt) {
        float s7 = (nt & 8) ? -1.f : 1.f;
        int ncol = nt * 16 + l15;
        const float* krBase = KTr + (size_t)ncol * DIM;  // per-lane contiguous row
        const float* kiBase = KTi + (size_t)ncol * DIM;
        v8f cr0 = {}, ci0 = {}, cr1 = {}, ci1 = {};
        #pragma unroll 4
        for (int k = 0; k < DIM; k += 4) {
            int krow = k + half * 2;
            // A frags (16x4 f32) for both row tiles: lane l15 = row M
            v2f ar0 = *(const v2f*)&lAr[l15 * ASTRIDE + krow];
            v2f ai0 = *(const v2f*)&lAi[l15 * ASTRIDE + krow];
            v2f ar1 = *(const v2f*)&lAr[(l15 + 16) * ASTRIDE + krow];
            v2f ai1 = *(const v2f*)&lAi[(l15 + 16) * ASTRIDE + krow];
            // B frag (4x16 f32): contiguous per-lane b64 thanks to KT layout
            v2f kr = *(const v2f*)&krBase[krow];
            v2f ki = *(const v2f*)&kiBase[krow];
            v2f nai0 = -ai0;  // f32 WMMA has no A/B NEG modifier; negate in VALU
            v2f nai1 = -ai1;
            // Cr += Ar*Kr - Ai*Ki ; Ci += Ar*Ki + Ai*Kr   (x2 row tiles)
            cr0 = __builtin_amdgcn_wmma_f32_16x16x4_f32(false, ar0,  false, kr, (short)0, cr0, false, false);
            cr0 = __builtin_amdgcn_wmma_f32_16x16x4_f32(false, nai0, false, ki, (short)0, cr0, false, false);
            ci0 = __builtin_amdgcn_wmma_f32_16x16x4_f32(false, ar0,  false, ki, (short)0, ci0, false, false);
            ci0 = __builtin_amdgcn_wmma_f32_16x16x4_f32(false, ai0,  false, kr, (short)0, ci0, false, false);
            cr1 = __builtin_amdgcn_wmma_f32_16x16x4_f32(false, ar1,  false, kr, (short)0, cr1, false, false);
            cr1 = __builtin_amdgcn_wmma_f32_16x16x4_f32(false, nai1, false, ki, (short)0, cr1, false, false);
            ci1 = __builtin_amdgcn_wmma_f32_16x16x4_f32(false, ar1,  false, ki, (short)0, ci1, false, false);
            ci1 = __builtin_amdgcn_wmma_f32_16x16x4_f32(false, ai1,  false, kr, (short)0, ci1, false, false);
        }
        #pragma unroll
        for (int r = 0; r < 8; ++r) {
            float p0 = cr0[r] * cr0[r] + ci0[r] * ci0[r];
            float p1 = cr1[r] * cr1[r] + ci1[r] * ci1[r];
            acc3[0][r] += s3 * p0;  acc7[0][r] += s7 * p0;
            acc3[1][r] += s3 * p1;  acc7[1][r] += s7 * p1;
        }
    }

    // reduce over N (16 lanes within each half; masks 1..8 never cross halves)
    #pragma unroll
    for (int m = 1; m <= 8; m <<= 1) {
        #pragma unroll
        for (int tfrag = 0; tfrag < 2; ++tfrag)
            #pragma unroll
            for (int r = 0; r < 8; ++r) {
                acc3[tfrag][r] += __shfl_xor(acc3[tfrag][r], m, 32);
                acc7[tfrag][r] += __shfl_xor(acc7[tfrag][r], m, 32);
            }
    }
    if (l15 == 0) {
        #pragma unroll
        for (int tfrag = 0; tfrag < 2; ++tfrag)
            #pragma unroll
            for (int r = 0; r < 8; ++r) {
                int row = r0 + tfrag * 16 + half * 8 + r;  // D: VGPR r -> M=r (+8 upper half)
                feats[row * 6 + 2 * br + 0] = acc3[tfrag][r];
                feats[row * 6 + 2 * br + 1] = acc7[tfrag][r];
            }
    }
}

// ------------------------------ Phase 4: MLP -------------------------------
__global__ void mlp_kernel(const float* __restrict__ feats,
                           const float* __restrict__ W1, const float* __restrict__ b1,
                           const float* __restrict__ W2, const float* __restrict__ b2,
                           float* __restrict__ out, int B) {
    int b = blockIdx.x * blockDim.x + threadIdx.x;
    if (b >= B) return;
    float f[6];
    #pragma unroll
    for (int i = 0; i < 6; ++i) f[i] = feats[b * 6 + i];
    float acc = b2[0];
    #pragma unroll
    for (int j = 0; j < 12; ++j) {
        float h = b1[j];
        #pragma unroll
        for (int i = 0; i < 6; ++i) h += f[i] * W1[j * 6 + i];
        acc += tanhf(h) * W2[j];
    }
    out[b] = 1.f / (1.f + expf(-acc));
}

// ---------------------------------------------------------------------------
extern "C" void kernel_launch(void* const* d_in, const int* in_sizes, int n_in,
                              void* d_out, int out_size, void* d_ws, size_t ws_size,
                              hipStream_t stream) {
    (void)n_in; (void)out_size; (void)ws_size;
    const float* theta = (const float*)d_in[0];
    const float* phi   = (const float*)d_in[1];
    const float* ax    = (const float*)d_in[2];
    const float* ay    = (const float*)d_in[3];
    const float* az    = (const float*)d_in[4];
    const float* u3x   = (const float*)d_in[5];
    const float* u3y   = (const float*)d_in[6];
    const float* u3z   = (const float*)d_in[7];
    const float* W1    = (const float*)d_in[8];
    const float* b1    = (const float*)d_in[9];
    const float* W2    = (const float*)d_in[10];
    const float* b2    = (const float*)d_in[11];

    const int B = in_sizes[0];                 // 8192
    float* ws    = (float*)d_ws;
    float* Ar    = ws;                         // B*256
    float* Ai    = Ar + (size_t)B * DIM;       // B*256
    float* K     = Ai + (size_t)B * DIM;       // 3 * 2 * 256*256 (transposed planes)
    float* feats = K + (size_t)3 * 2 * DIM * DIM;   // B*6

    // Phase 1: one wave per batch state (8 waves / 256-thread block)
    init_kernel<<<(B + 7) / 8, 256, 0, stream>>>(theta, phi, Ar, Ai, B);
    // Phase 2: 768 basis-state simulations -> KT planes
    branch_kernel<<<96, 256, 0, stream>>>(ax, ay, az, u3x, u3y, u3z, K);
    // Phase 3: WMMA complex GEMM + fused <Z3>,<Z7>
    int rowBlocks = B / ROWS;
    gemm_expz_kernel<<<rowBlocks * 3, 32, 0, stream>>>(Ar, Ai, K, feats, rowBlocks);
    // Phase 4: MLP head
    mlp_kernel<<<(B + 255) / 256, 256, 0, stream>>>(feats, W1, b1, W2, b2, (float*)d_out, B);
}